// SparseDAG_36825049596267
// MI455X (gfx1250) — compile-verified
//
#include <hip/hip_runtime.h>
#if __has_include(<hip/amd_detail/amd_gfx1250_TDM.h>)
#define TDM_6ARG 1
#else
#define TDM_6ARG 0
#endif
#include <cstddef>
#include <cstdint>

constexpr int N_NODES = 8192;
constexpr int BLK     = 256;             // columns per sequential block
constexpr int NB      = N_NODES / BLK;   // 32 blocks
constexpr int NWG     = 128;             // persistent workgroups
constexpr int TPB     = 256;             // 8 wave32 per WG
constexpr int NWAVES  = NWG * (TPB / 32);
constexpr int PF      = 8;               // tri-phase row prefetch depth
constexpr int DBAND   = 4;               // register band depth
constexpr int TROWS   = 32;              // rows per TDM subtile (16 cols -> 2KB)
constexpr int SUBS    = 8;               // subtiles per task (256 rows)

typedef float        v2f __attribute__((ext_vector_type(2)));
typedef float        v8f __attribute__((ext_vector_type(8)));
typedef unsigned int v4u __attribute__((ext_vector_type(4)));
typedef int          v4i __attribute__((ext_vector_type(4)));
typedef int          v8i __attribute__((ext_vector_type(8)));

// Issue one TDM tile load: 2D tile, 16 cols x TROWS rows, fp32, row stride N.
__device__ __forceinline__ void tdm_load_tile(unsigned lds_byte,
                                              unsigned long long gaddr,
                                              v8i g1) {
    // D# group0: count=1 | lds_addr | global_addr[56:0] | type=2
    v4u g0 = { 1u, lds_byte, (unsigned)gaddr,
               ((unsigned)(gaddr >> 32) & 0x01FFFFFFu) | 0x80000000u };
    v4i gz = { 0, 0, 0, 0 };
#if TDM_6ARG
    v8i gz8 = { 0, 0, 0, 0, 0, 0, 0, 0 };
    __builtin_amdgcn_tensor_load_to_lds(g0, g1, gz, gz, gz8, 0);
#else
    __builtin_amdgcn_tensor_load_to_lds(g0, g1, gz, gz, 0);
#endif
}

__global__ void dag_init(const float* __restrict__ x, const float* __restrict__ b,
                         float* s, float* acc, int* ctrl) {
    int i = blockIdx.x * blockDim.x + threadIdx.x;
    if (i < N_NODES) {
        acc[i] = b[i];
        s[i]   = (i == 0) ? x[0] : (i == 1) ? x[1] : 0.0f;
    }
    if (i == 0) { ctrl[0] = 0; ctrl[1] = 0; }            // arrive, release
}

__global__ void __launch_bounds__(TPB, 1)
dag_scan(const float* __restrict__ x, const float* __restrict__ W,
         float* s, float* acc, int* ctrl, float* out) {
    __shared__ volatile float t_sh[BLK];                        // scan state
    __shared__ __attribute__((aligned(16)))
        float stage[TPB / 32][2][TROWS * 16];                   // TDM staging, 32KB
    const int lane     = threadIdx.x & 31;
    const int wave_gid = blockIdx.x * (TPB / 32) + (threadIdx.x >> 5);
    int* arrive  = ctrl;
    int* release = ctrl + 1;

    // D# group1: mask=0, data_size=4B, tensor 16x32 (== tile), tile 16x32,
    // dim0 stride = 8192 elements.  Bits per CDNA5 ISA.
    const v8i g1 = { 0x00020000, 0x00100000, 0x00200000, 0x00100000,
                     0x00000020, 0x00002000, 0x00000000, 0x00000000 };

    for (int k = 0; k < NB; ++k) {
        const int jcol0  = k * BLK;
        // ---------- dense panel via TDM -> LDS -> WMMA ----------
        const int ntasks = 16 * k;                       // 16 strips x k chunks(256 rows)
        for (int tid = wave_gid; tid < ntasks; tid += NWAVES) {
            const int strip = tid & 15;
            const int jc    = jcol0 + strip * 16;
            // wave-uniform scalars for the descriptor (must live in SGPRs)
            const int wslot = __builtin_amdgcn_readfirstlane(threadIdx.x >> 5);
            const int jcu   = __builtin_amdgcn_readfirstlane(jc);
            const int rbase = __builtin_amdgcn_readfirstlane((tid >> 4) << 8);
            const unsigned lds0 = (unsigned)(uintptr_t)&stage[wslot][0][0];
            const unsigned lds1 = (unsigned)(uintptr_t)&stage[wslot][1][0];
            const unsigned long long wbase = (unsigned long long)(uintptr_t)W
                + ((unsigned long long)rbase * N_NODES + (unsigned)jcu) * 4ull;
            const unsigned long long substep = (unsigned long long)TROWS * N_NODES * 4ull;

            tdm_load_tile(lds0, wbase, g1);
            tdm_load_tile(lds1, wbase + substep, g1);

            v8f c = {0.f,0.f,0.f,0.f,0.f,0.f,0.f,0.f};
            const int n  = lane & 15;
            const int kp = (lane >> 4) << 1;
            for (int sub = 0; sub < SUBS; ++sub) {
                if (sub < SUBS - 1) __builtin_amdgcn_s_wait_tensorcnt((short)1);
                else                __builtin_amdgcn_s_wait_tensorcnt((short)0);
                const int buf = sub & 1;
                const int i4b = rbase + sub * TROWS;
                #pragma unroll
                for (int ii = 0; ii < TROWS / 4; ++ii) {
                    const int r4 = ii * 4;
                    v2f sa, wb;
                    sa.x = s[i4b + r4 + kp];                     // A: s broadcast
                    sa.y = s[i4b + r4 + kp + 1];
                    wb.x = stage[wslot][buf][(r4 + kp)     * 16 + n];  // B: W tile (LDS)
                    wb.y = stage[wslot][buf][(r4 + kp + 1) * 16 + n];
                    c = __builtin_amdgcn_wmma_f32_16x16x4_f32(
                            false, sa, false, wb, (short)0, c, false, false);
                }
                if (sub + 2 < SUBS)                              // refill this buffer
                    tdm_load_tile(buf ? lds1 : lds0,
                                  wbase + (unsigned long long)(sub + 2) * substep, g1);
            }
            if (lane < 16) atomicAdd(&acc[jc + lane], c[0]);     // D row 0, col = lane
        }

        // ---------- arrive (one per WG) ----------
        __threadfence();
        if (threadIdx.x == 0) atomicAdd(arrive, 1);

        if (blockIdx.x == 0 && threadIdx.x < 32) {
            while (__hip_atomic_load(arrive, __ATOMIC_RELAXED, __HIP_MEMORY_SCOPE_AGENT)
                   < NWG * (k + 1))
                __builtin_amdgcn_s_sleep(1);
            __threadfence();

            const float x0 = x[0], x1 = x[1];

            #pragma unroll
            for (int r = 0; r < 8; ++r)
                t_sh[lane + 32 * r] = acc[jcol0 + lane + 32 * r];

            float wrow[PF][8];
            float sdp[PF][DBAND];
            #pragma unroll
            for (int p = 0; p < PF; ++p) {
                const float* rw = W + (size_t)(jcol0 + p) * N_NODES;
                #pragma unroll
                for (int r = 0; r < 8; ++r) wrow[p][r] = rw[jcol0 + lane + 32 * r];
                #pragma unroll
                for (int d = 0; d < DBAND; ++d) {
                    int col = jcol0 + p + 1 + d;
                    sdp[p][d] = rw[col < N_NODES ? col : (N_NODES - 1)];
                }
            }

            float u0 = t_sh[0], u1 = t_sh[1], u2 = t_sh[2], u3 = t_sh[3], u4 = t_sh[4];

            for (int base = 0; base < BLK; base += PF) {
                #pragma unroll
                for (int p = 0; p < PF; ++p) {
                    const int joff = base + p;
                    float sj;
                    if (k == 0 && joff < 2) sj = (joff == 0) ? x0 : x1;
                    else                    sj = 1.0f / (1.0f + __expf(-u0));
                    if (lane == 0) {
                        s[jcol0 + joff] = sj;
                        if (k == NB - 1 && joff == BLK - 1) out[0] = sj;
                    }
                    u0 = u1 + sdp[p][0] * sj;                   // critical chain
                    u1 = u2 + sdp[p][1] * sj;
                    u2 = u3 + sdp[p][2] * sj;
                    u3 = u4 + sdp[p][3] * sj;
                    #pragma unroll
                    for (int r = 0; r < 8; ++r) {
                        const int jj = lane + 32 * r;
                        if (jj > joff + DBAND) t_sh[jj] += wrow[p][r] * sj;
                    }
                    {
                        int ld = joff + 1 + DBAND;
                        u4 = t_sh[ld < BLK ? ld : (BLK - 1)];
                    }
                    const int nj = joff + PF;
                    if (nj < BLK) {
                        const float* rw = W + (size_t)(jcol0 + nj) * N_NODES;
                        #pragma unroll
                        for (int r = 0; r < 8; ++r)
                            wrow[p][r] = rw[jcol0 + lane + 32 * r];
                        #pragma unroll
                        for (int d = 0; d < DBAND; ++d) {
                            int col = jcol0 + nj + 1 + d;
                            sdp[p][d] = rw[col < N_NODES ? col : (N_NODES - 1)];
                        }
                    }
                }
            }
            __threadfence();
            if (lane == 0)
                __hip_atomic_store(release, k + 1, __ATOMIC_RELAXED,
                                   __HIP_MEMORY_SCOPE_AGENT);
        } else {
            while (__hip_atomic_load(release, __ATOMIC_RELAXED, __HIP_MEMORY_SCOPE_AGENT)
                   < k + 1)
                __builtin_amdgcn_s_sleep(1);
            __threadfence();
        }
    }
}

extern "C" void kernel_launch(void* const* d_in, const int* in_sizes, int n_in,
                              void* d_out, int out_size, void* d_ws, size_t ws_size,
                              hipStream_t stream) {
    const float* x = (const float*)d_in[0];   // (2,)
    const float* W = (const float*)d_in[1];   // (N,N) row-major
    const float* b = (const float*)d_in[2];   // (N,)
    float* ws  = (float*)d_ws;
    float* s   = ws;                          // N floats
    float* acc = ws + N_NODES;                // N floats
    int*  ctrl = (int*)(ws + 2 * N_NODES);    // arrive, release
    float* out = (float*)d_out;

    dag_init<<<N_NODES / TPB, TPB, 0, stream>>>(x, b, s, acc, ctrl);
    dag_scan<<<NWG, TPB, 0, stream>>>(x, W, s, acc, ctrl, out);
}